// LSTMModel_18691697672685
// MI455X (gfx1250) — compile-verified
//
#include <hip/hip_runtime.h>

typedef float v2f __attribute__((ext_vector_type(2)));
typedef float v8f __attribute__((ext_vector_type(8)));

#define B_TOT 512
#define T_LEN 2048
#define I_LEN 7
#define H_LEN 64
#define G_LEN 256          // 4*H
#define K_TOT 72           // H + I + 1 (augmented K: [h | x | 1])
#define K_CH  18           // K_TOT / 4
#define SROW  76           // padded LDS row stride (floats) to avoid bank conflicts
#define SBUF_SZ (16 * SROW)

// gfx1250 has a native v_tanh_f32 transcendental; fall back to exp if the
// builtin isn't declared by this toolchain.
__device__ __forceinline__ float tanh_f(float x) {
#if __has_builtin(__builtin_amdgcn_tanhf)
    return __builtin_amdgcn_tanhf(x);
#else
    return 2.0f / (1.0f + __expf(-2.0f * x)) - 1.0f;
#endif
}
__device__ __forceinline__ float sigm(float x) {
    // sigmoid(x) = 0.5*tanh(x/2) + 0.5  (1 trans + 1 mul + 1 fma)
    return __builtin_fmaf(0.5f, tanh_f(0.5f * x), 0.5f);
}

// Wcat[k][g]:  k<64 -> W_hh[g][k]   (recurrent weights, transposed)
//              64<=k<71 -> W_ih[g][k-64]  (input weights, transposed)
//              k==71 -> b_ih[g] + b_hh[g] (fused bias row; state col is constant 1)
__device__ __forceinline__ float wcat_ld(const float* __restrict__ W_ih,
                                         const float* __restrict__ W_hh,
                                         const float* __restrict__ b_ih,
                                         const float* __restrict__ b_hh,
                                         int g, int k) {
    if (k < H_LEN)               return W_hh[g * H_LEN + k];
    else if (k < H_LEN + I_LEN)  return W_ih[g * I_LEN + (k - H_LEN)];
    else                         return b_ih[g] + b_hh[g];
}

// One LSTM direction over [t_start, t_start + t_count) (ascending; for the
// backward direction only the first step matters, so t_count==1 at t=T-1).
// Each workgroup handles 16 batch rows; 4 waves each own a 16-column slice
// of the gates: wave w computes tiles i/f/g/o at column offset w*16, so the
// elementwise LSTM cell update happens entirely in registers.
// State buffer is double-buffered in LDS -> ONE barrier per timestep.
__global__ __launch_bounds__(128, 1) void lstm_scan_kernel(
    const float* __restrict__ x,      // [B, T, I]
    const float* __restrict__ W_ih,   // [4H, I]
    const float* __restrict__ W_hh,   // [4H, H]
    const float* __restrict__ b_ih,   // [4H]
    const float* __restrict__ b_hh,   // [4H]
    float* __restrict__ h_out,        // [B, H]
    int t_start, int t_count)
{
    __shared__ float sbuf[2 * SBUF_SZ]; // 2 x per-row [h(64) | x(7) | 1 | pad(4)]

    const int tid  = threadIdx.x;
    const int wave = tid >> 5;        // 0..3
    const int lane = tid & 31;
    const int l16  = lane & 15;
    const int hi   = lane >> 4;       // 0: K+{0,1}, rows M;  1: K+{2,3}, rows M (A) / M+8 (C/D)
    const int b0   = blockIdx.x * 16; // batch tile base
    const int wcol = wave * 16 + l16; // this lane's gate column within a 64-wide gate

    // ---- Preload loop-invariant B fragments (weights) into VGPRs ----
    // B-matrix (KxN) fragment layout for V_WMMA_F32_16X16X4_F32:
    //   lanes 0-15:  N=lane,    vgpr0=K0, vgpr1=K1
    //   lanes 16-31: N=lane-16, vgpr0=K2, vgpr1=K3
    v2f bf0[K_CH], bf1[K_CH], bf2[K_CH], bf3[K_CH];
#pragma unroll
    for (int kc = 0; kc < K_CH; kc++) {
        const int k = kc * 4 + hi * 2;
        bf0[kc].x = wcat_ld(W_ih, W_hh, b_ih, b_hh, wcol + 0 * H_LEN, k);
        bf0[kc].y = wcat_ld(W_ih, W_hh, b_ih, b_hh, wcol + 0 * H_LEN, k + 1);
        bf1[kc].x = wcat_ld(W_ih, W_hh, b_ih, b_hh, wcol + 1 * H_LEN, k);
        bf1[kc].y = wcat_ld(W_ih, W_hh, b_ih, b_hh, wcol + 1 * H_LEN, k + 1);
        bf2[kc].x = wcat_ld(W_ih, W_hh, b_ih, b_hh, wcol + 2 * H_LEN, k);
        bf2[kc].y = wcat_ld(W_ih, W_hh, b_ih, b_hh, wcol + 2 * H_LEN, k + 1);
        bf3[kc].x = wcat_ld(W_ih, W_hh, b_ih, b_hh, wcol + 3 * H_LEN, k);
        bf3[kc].y = wcat_ld(W_ih, W_hh, b_ih, b_hh, wcol + 3 * H_LEN, k + 1);
    }

    // ---- Initialize buffer 0: h0 = 0, constant-1 column in BOTH buffers,
    //      x[t_start] into buffer 0. Buffer 1's h/x regions are fully written
    //      by iteration 0 before they are ever read.
    for (int idx = tid; idx < SBUF_SZ; idx += 128) sbuf[idx] = 0.0f;
    __syncthreads();
    if (tid < 16) {
        sbuf[tid * SROW + (K_TOT - 1)]           = 1.0f;
        sbuf[SBUF_SZ + tid * SROW + (K_TOT - 1)] = 1.0f;
    }
    const int    xm    = tid / 7;                 // batch row within tile (tid<112)
    const int    xi    = tid % 7;                 // input feature
    const size_t xbase = (size_t)(b0 + xm) * ((size_t)T_LEN * I_LEN) + xi;
    if (tid < 112) sbuf[xm * SROW + H_LEN + xi] = x[xbase + (size_t)t_start * I_LEN];

    v8f c_val = {};   // cell state tile, lives in registers for all 2048 steps
    v8f h_new = {};

    for (int s = 0; s < t_count; s++) {
        const int t = t_start + s;
        float* __restrict__ cur = &sbuf[(s & 1) * SBUF_SZ];        // read h_t, x_t
        float* __restrict__ nxt = &sbuf[((s + 1) & 1) * SBUF_SZ];  // write h_{t+1}, x_{t+1}

        __syncthreads();   // prior iteration's writes to `cur` are visible

        // Prefetch next timestep's input while the WMMAs run
        float xnext = 0.0f;
        if ((s + 1 < t_count) && (tid < 112))
            xnext = x[xbase + (size_t)(t + 1) * I_LEN];

        // gates[16x256] = s[16x72] @ Wcat[72x256], K in chunks of 4
        v8f acc0 = {}, acc1 = {}, acc2 = {}, acc3 = {};
        const int arow = l16 * SROW + hi * 2;
#pragma unroll
        for (int kc = 0; kc < K_CH; kc++) {
            // A-matrix (16x4) fragment: lanes 0-15 row=lane K{0,1}; lanes 16-31 row=lane-16 K{2,3}
            v2f a;
            a.x = cur[arow + kc * 4];
            a.y = cur[arow + kc * 4 + 1];
            acc0 = __builtin_amdgcn_wmma_f32_16x16x4_f32(false, a, false, bf0[kc], (short)0, acc0, false, false);
            acc1 = __builtin_amdgcn_wmma_f32_16x16x4_f32(false, a, false, bf1[kc], (short)0, acc1, false, false);
            acc2 = __builtin_amdgcn_wmma_f32_16x16x4_f32(false, a, false, bf2[kc], (short)0, acc2, false, false);
            acc3 = __builtin_amdgcn_wmma_f32_16x16x4_f32(false, a, false, bf3[kc], (short)0, acc3, false, false);
        }

        // LSTM cell update, fully in registers (acc0..3 share the same (m,n) map)
#pragma unroll
        for (int r = 0; r < 8; r++) {
            const float ig = sigm(acc0[r]);
            const float fg = sigm(acc1[r]);
            const float gg = tanh_f(acc2[r]);
            const float og = sigm(acc3[r]);
            const float c  = __builtin_fmaf(fg, c_val[r], ig * gg);
            c_val[r] = c;
            h_new[r] = og * tanh_f(c);
        }

        // Publish into the OTHER buffer; no barrier needed before these writes.
        // C/D layout -> vgpr r = row (r + hi*8), col = wcol
#pragma unroll
        for (int r = 0; r < 8; r++)
            nxt[(r + hi * 8) * SROW + wcol] = h_new[r];
        if ((s + 1 < t_count) && (tid < 112))
            nxt[xm * SROW + H_LEN + xi] = xnext;
    }

    // Final hidden state to global
#pragma unroll
    for (int r = 0; r < 8; r++)
        h_out[(size_t)(b0 + r + hi * 8) * H_LEN + wcol] = h_new[r];
}

// out[b,o] = concat(hf[b], hb[b]) . W_lin[o] + b_lin[o]   (O=3, tiny)
__global__ __launch_bounds__(128) void head_kernel(
    const float* __restrict__ hf,     // [B, H]
    const float* __restrict__ hb,     // [B, H]
    const float* __restrict__ W_lin,  // [3, 2H]
    const float* __restrict__ b_lin,  // [3]
    float* __restrict__ out)          // [B, 3]
{
    const int b = blockIdx.x * blockDim.x + threadIdx.x;
    if (b >= B_TOT) return;
    float a0 = b_lin[0], a1 = b_lin[1], a2 = b_lin[2];
#pragma unroll 4
    for (int j = 0; j < H_LEN; j++) {
        const float f = hf[b * H_LEN + j];
        const float r = hb[b * H_LEN + j];
        a0 += f * W_lin[0 * 2 * H_LEN + j] + r * W_lin[0 * 2 * H_LEN + H_LEN + j];
        a1 += f * W_lin[1 * 2 * H_LEN + j] + r * W_lin[1 * 2 * H_LEN + H_LEN + j];
        a2 += f * W_lin[2 * 2 * H_LEN + j] + r * W_lin[2 * 2 * H_LEN + H_LEN + j];
    }
    out[b * 3 + 0] = a0;
    out[b * 3 + 1] = a1;
    out[b * 3 + 2] = a2;
}

extern "C" void kernel_launch(void* const* d_in, const int* in_sizes, int n_in,
                              void* d_out, int out_size, void* d_ws, size_t ws_size,
                              hipStream_t stream) {
    const float* x    = (const float*)d_in[0];
    const float* Wihf = (const float*)d_in[1];
    const float* Whhf = (const float*)d_in[2];
    const float* bihf = (const float*)d_in[3];
    const float* bhhf = (const float*)d_in[4];
    const float* Wihb = (const float*)d_in[5];
    const float* Whhb = (const float*)d_in[6];
    const float* bihb = (const float*)d_in[7];
    const float* bhhb = (const float*)d_in[8];
    const float* Wlin = (const float*)d_in[9];
    const float* blin = (const float*)d_in[10];
    float* out = (float*)d_out;

    float* hf = (float*)d_ws;            // [512, 64]
    float* hb = hf + B_TOT * H_LEN;      // [512, 64]

    // Forward: full sequential scan over T=2048.
    lstm_scan_kernel<<<B_TOT / 16, 128, 0, stream>>>(x, Wihf, Whhf, bihf, bhhf, hf, 0, T_LEN);
    // Backward direction's time-aligned value at t=T-1 is its FIRST step:
    // one LSTM step on x[T-1] with zero initial state.
    lstm_scan_kernel<<<B_TOT / 16, 128, 0, stream>>>(x, Wihb, Whhb, bihb, bhhb, hb, T_LEN - 1, 1);
    // Tiny output head.
    head_kernel<<<(B_TOT + 127) / 128, 128, 0, stream>>>(hf, hb, Wlin, blin, out);
}